// EncoderSpin_28080496181856
// MI455X (gfx1250) — compile-verified
//
#include <hip/hip_runtime.h>

#define N_NODES 50000
#define N_EDGES 1600000
#define N_GRAPHS 50
#define ETOT (N_EDGES + N_NODES)

typedef __attribute__((ext_vector_type(2))) float v2f;
typedef __attribute__((ext_vector_type(8))) float v8f;

static inline int cdiv(long long a, int b) { return (int)((a + b - 1) / b); }

// ---------- monotone float<->uint encoding for atomic max ----------
__device__ __forceinline__ unsigned enc_f(float f) {
  unsigned u = __float_as_uint(f);
  return (u & 0x80000000u) ? ~u : (u | 0x80000000u);
}
__device__ __forceinline__ float dec_f(unsigned u) {
  return (u & 0x80000000u) ? __uint_as_float(u & 0x7FFFFFFFu) : __uint_as_float(~u);
}

// ---------- utility kernels ----------
__global__ void k_zero(float* p, long long n) {
  long long i = blockIdx.x * (long long)blockDim.x + threadIdx.x;
  if (i < n) p[i] = 0.f;
}

__global__ void k_zero4(float4* p, long long n4) {
  long long i = blockIdx.x * (long long)blockDim.x + threadIdx.x;
  if (i < n4) p[i] = make_float4(0.f, 0.f, 0.f, 0.f);
}

__global__ void k_mean_reduce(const float* __restrict__ w, float* out, int n) {
  __shared__ float sm[256];
  float s = 0.f;
  for (long long i = blockIdx.x * (long long)blockDim.x + threadIdx.x; i < n;
       i += (long long)gridDim.x * blockDim.x)
    s += w[i];
  sm[threadIdx.x] = s;
  __syncthreads();
  for (int off = 128; off > 0; off >>= 1) {
    if ((int)threadIdx.x < off) sm[threadIdx.x] += sm[threadIdx.x + off];
    __syncthreads();
  }
  if (threadIdx.x == 0) atomicAdd(out, sm[0] * (1.0f / n));
}

__global__ void k_dot_small(const float* __restrict__ a, const float* __restrict__ b,
                            float* out, int n) {
  __shared__ float sm[64];
  int t = threadIdx.x;
  sm[t] = (t < n) ? a[t] * b[t] : 0.f;
  __syncthreads();
  for (int off = 32; off > 0; off >>= 1) {
    if (t < off) sm[t] += sm[t + off];
    __syncthreads();
  }
  if (t == 0) out[0] = sm[0];
}

// ---------- dense lift: h = x @ lin ----------
template <int DOUT>
__global__ void k_lin1(const float* __restrict__ x, const float* __restrict__ lin,
                       float* __restrict__ H) {
  long long gid = blockIdx.x * (long long)blockDim.x + threadIdx.x;
  if (gid >= (long long)N_NODES * DOUT) return;
  int i = (int)(gid / DOUT), j = (int)(gid % DOUT);
  H[gid] = x[i] * lin[j];
}

// f32 WMMA GEMM: one wave per 16x16 C tile, V_WMMA_F32_16X16X4_F32 along K.
// A (16x4 f32): lane L -> M = L%16, K = v + 2*(L/16) for VGPR v.
// B (4x16 f32): lane L -> N = L%16, K = v + 2*(L/16).
// C/D (16x16):  VGPR r, lane L -> M = r + 8*(L/16), N = L%16.
template <int K, int DOUT>
__global__ void k_gemm_wmma(const float* __restrict__ X, const float* __restrict__ Wm,
                            float* __restrict__ H) {
  const int TN = DOUT / 16;
  const int total = (N_NODES / 16) * TN;
  int wv = (int)((blockIdx.x * (long long)blockDim.x + threadIdx.x) >> 5);
  int lane = threadIdx.x & 31;
  if (wv >= total) return;  // wave-uniform: EXEC stays all-ones for WMMA
  int mt = wv / TN, nt = wv % TN;
  int lm = lane & 15;
  int lh = lane >> 4;
  const float* xrow = X + (size_t)(mt * 16 + lm) * K;
  const float* wcol = Wm + nt * 16 + lm;
  v8f acc = {};
#pragma unroll
  for (int k0 = 0; k0 < K; k0 += 4) {
    int ka = k0 + 2 * lh;
    v2f a, b;
    a.x = xrow[ka];
    a.y = xrow[ka + 1];
    b.x = wcol[(size_t)ka * DOUT];
    b.y = wcol[(size_t)(ka + 1) * DOUT];
    acc = __builtin_amdgcn_wmma_f32_16x16x4_f32(false, a, false, b, (short)0, acc,
                                                false, false);
  }
  float* orow = H + (size_t)(mt * 16 + 8 * lh) * DOUT + nt * 16 + lm;
#pragma unroll
  for (int r = 0; r < 8; ++r) orow[(size_t)r * DOUT] = acc[r];
}

// ---------- per-node attention scores ----------
template <int D>
__global__ void k_node_scores(const float* __restrict__ H, const float* __restrict__ asrc,
                              const float* __restrict__ adst, float* __restrict__ as_,
                              float* __restrict__ ad_) {
  int wv = (int)((blockIdx.x * (long long)blockDim.x + threadIdx.x) >> 5);
  int lane = threadIdx.x & 31;
  if (wv >= N_NODES) return;
  float s = 0.f, d = 0.f;
  for (int j = lane; j < D; j += 32) {
    float v = H[(size_t)wv * D + j];
    s += v * asrc[j];
    d += v * adst[j];
  }
  for (int off = 16; off > 0; off >>= 1) {
    s += __shfl_down(s, off, 32);
    d += __shfl_down(d, off, 32);
  }
  if (lane == 0) { as_[wv] = s; ad_[wv] = d; }
}

// ---------- edge phase ----------
__global__ void k_edge_score_max(const int* __restrict__ ei, const float* __restrict__ ew,
                                 const float* __restrict__ scal,  // [0]=mean_ew, [1]=c
                                 const float* __restrict__ as_, const float* __restrict__ ad_,
                                 float* __restrict__ aex, unsigned* __restrict__ menc) {
  int e = blockIdx.x * blockDim.x + threadIdx.x;
  if (e >= ETOT) return;
  int s, d;
  float ea;
  if (e < N_EDGES) { s = ei[e]; d = ei[N_EDGES + e]; ea = ew[e]; }
  else { s = d = e - N_EDGES; ea = scal[0]; }
  float a = as_[s] + ad_[d] + scal[1] * ea;
  a = (a > 0.f) ? a : 0.2f * a;  // leaky_relu(0.2)
  aex[e] = a;
  atomicMax(menc + d, enc_f(a));
}

__global__ void k_edge_exp_sum(const int* __restrict__ ei, float* __restrict__ aex,
                               const unsigned* __restrict__ menc, float* __restrict__ ssum) {
  int e = blockIdx.x * blockDim.x + threadIdx.x;
  if (e >= ETOT) return;
  int d = (e < N_EDGES) ? ei[N_EDGES + e] : (e - N_EDGES);
  float ex = __expf(aex[e] - dec_f(menc[d]));
  aex[e] = ex;
  atomicAdd(ssum + d, ex);
}

// normalize softmax weight once per edge (hoisted out of the D-wide scatter)
__global__ void k_edge_norm(const int* __restrict__ ei, float* __restrict__ aex,
                            const float* __restrict__ ssum) {
  int e = blockIdx.x * blockDim.x + threadIdx.x;
  if (e >= ETOT) return;
  int d = (e < N_EDGES) ? ei[N_EDGES + e] : (e - N_EDGES);
  aex[e] = aex[e] / ssum[d];
}

// weighted scatter-add: thread = (edge, 4-channel chunk); float4 gather of h[src]
template <int D>
__global__ void k_aggregate(const int* __restrict__ ei, const float* __restrict__ wnorm,
                            const float* __restrict__ H, float* __restrict__ out) {
  const int C = D / 4;
  long long gid = blockIdx.x * (long long)blockDim.x + threadIdx.x;
  if (gid >= (long long)ETOT * C) return;
  int e = (int)(gid / C);
  int jc = (int)(gid % C) * 4;
  int s, d;
  if (e < N_EDGES) { s = ei[e]; d = ei[N_EDGES + e]; }
  else { s = d = e - N_EDGES; }
  float w = wnorm[e];
  float4 h = *(const float4*)(H + (size_t)s * D + jc);
  float* o = out + (size_t)d * D + jc;
  atomicAdd(o + 0, w * h.x);
  atomicAdd(o + 1, w * h.y);
  atomicAdd(o + 2, w * h.z);
  atomicAdd(o + 3, w * h.w);
}

template <int D>
__global__ void k_bias_act(float* __restrict__ x, const float* __restrict__ bias, int relu) {
  const int C = D / 4;
  long long gid = blockIdx.x * (long long)blockDim.x + threadIdx.x;
  if (gid >= (long long)N_NODES * C) return;
  int jc = (int)(gid % C) * 4;
  float4 b = *(const float4*)(bias + jc);
  float4 v = ((float4*)x)[gid];
  v.x += b.x; v.y += b.y; v.z += b.z; v.w += b.w;
  if (relu) {
    v.x = fmaxf(v.x, 0.f); v.y = fmaxf(v.y, 0.f);
    v.z = fmaxf(v.z, 0.f); v.w = fmaxf(v.w, 0.f);
  }
  ((float4*)x)[gid] = v;
}

// ---------- batch norm ----------
template <int D>
__global__ void k_bn_reduce(const float* __restrict__ x, float* __restrict__ sum,
                            float* __restrict__ sq) {
  __shared__ float ssum[D], ssq[D];
  int tid = threadIdx.x;
  if (tid < D) { ssum[tid] = 0.f; ssq[tid] = 0.f; }
  __syncthreads();
  int j = tid % D;  // blockDim=256, D | 256 -> channel is loop-invariant
  float ls = 0.f, lq = 0.f;
  for (long long idx = blockIdx.x * (long long)blockDim.x + tid; idx < (long long)N_NODES * D;
       idx += (long long)gridDim.x * blockDim.x) {
    float v = x[idx];
    ls += v;
    lq += v * v;
  }
  atomicAdd(&ssum[j], ls);
  atomicAdd(&ssq[j], lq);
  __syncthreads();
  if (tid < D) { atomicAdd(&sum[tid], ssum[tid]); atomicAdd(&sq[tid], ssq[tid]); }
}

template <int D>
__global__ void k_bn_apply(float* __restrict__ x, const float* __restrict__ sum,
                           const float* __restrict__ sq, const float* __restrict__ gamma,
                           const float* __restrict__ beta) {
  const int C = D / 4;
  long long gid = blockIdx.x * (long long)blockDim.x + threadIdx.x;
  if (gid >= (long long)N_NODES * C) return;
  int jc = (int)(gid % C) * 4;
  float4 v = ((float4*)x)[gid];
  float* vp = &v.x;
#pragma unroll
  for (int t = 0; t < 4; ++t) {
    int j = jc + t;
    float mu = sum[j] * (1.0f / N_NODES);
    float var = fmaxf(sq[j] * (1.0f / N_NODES) - mu * mu, 0.f);
    vp[t] = gamma[j] * (vp[t] - mu) * rsqrtf(var + 1e-5f) + beta[j];
  }
  ((float4*)x)[gid] = v;
}

// ---------- graph norm (D = 64) ----------
__global__ void k_gn_cnt(const int* __restrict__ batch, float* __restrict__ cnt) {
  int i = blockIdx.x * blockDim.x + threadIdx.x;
  if (i < N_NODES) atomicAdd(&cnt[batch[i]], 1.0f);
}
__global__ void k_gn_mean(const float* __restrict__ x, const int* __restrict__ batch,
                          float* __restrict__ meanG) {
  long long gid = blockIdx.x * (long long)blockDim.x + threadIdx.x;
  if (gid >= (long long)N_NODES * 16) return;
  int i = (int)(gid >> 4), jc = (int)(gid & 15) * 4;
  float4 v = *(const float4*)(x + (size_t)i * 64 + jc);
  float* mg = &meanG[batch[i] * 64 + jc];
  atomicAdd(mg + 0, v.x);
  atomicAdd(mg + 1, v.y);
  atomicAdd(mg + 2, v.z);
  atomicAdd(mg + 3, v.w);
}
__global__ void k_gn_xcvar(const float* __restrict__ x, const int* __restrict__ batch,
                           const float* __restrict__ cnt, const float* __restrict__ meanG,
                           const float* __restrict__ mscale, float* __restrict__ xc,
                           float* __restrict__ varG) {
  long long gid = blockIdx.x * (long long)blockDim.x + threadIdx.x;
  if (gid >= (long long)N_NODES * 16) return;
  int i = (int)(gid >> 4), jc = (int)(gid & 15) * 4;
  int b = batch[i];
  float rcn = 1.0f / fmaxf(cnt[b], 1.f);
  float4 v = *(const float4*)(x + (size_t)i * 64 + jc);
  float* vp = &v.x;
  float* vg = &varG[b * 64 + jc];
#pragma unroll
  for (int t = 0; t < 4; ++t) {
    int j = jc + t;
    vp[t] = vp[t] - mscale[j] * (meanG[b * 64 + j] * rcn);
    atomicAdd(vg + t, vp[t] * vp[t]);
  }
  *(float4*)(xc + (size_t)i * 64 + jc) = v;
}
__global__ void k_gn_apply(const float* __restrict__ xc, const int* __restrict__ batch,
                           const float* __restrict__ cnt, const float* __restrict__ varG,
                           const float* __restrict__ wgt, const float* __restrict__ bias,
                           float* __restrict__ out) {
  long long gid = blockIdx.x * (long long)blockDim.x + threadIdx.x;
  if (gid >= (long long)N_NODES * 16) return;
  int i = (int)(gid >> 4), jc = (int)(gid & 15) * 4;
  int b = batch[i];
  float rcn = 1.0f / fmaxf(cnt[b], 1.f);
  float4 v = *(const float4*)(xc + (size_t)i * 64 + jc);
  float* vp = &v.x;
#pragma unroll
  for (int t = 0; t < 4; ++t) {
    int j = jc + t;
    vp[t] = wgt[j] * vp[t] * rsqrtf(varG[b * 64 + j] * rcn + 1e-5f) + bias[j];
  }
  *(float4*)(out + (size_t)i * 64 + jc) = v;
}

// ---------- per-layer driver ----------
template <int DIN, int DOUT>
static void run_gat(const float* xin, const float* lin, const float* att_src,
                    const float* att_dst, const float* lin_edge, const float* att_edge,
                    const float* bias, int relu, const int* ei, const float* ew, float* out,
                    float* bufH, float* aex, float* as_, float* ad_, unsigned* menc,
                    float* ssum, float* scal, hipStream_t stream) {
  const int TPB = 256;
  k_dot_small<<<1, 64, 0, stream>>>(lin_edge, att_edge, scal + 1, DOUT);
  if constexpr (DIN == 1) {
    k_lin1<DOUT><<<cdiv((long long)N_NODES * DOUT, TPB), TPB, 0, stream>>>(xin, lin, bufH);
  } else {
    int waves = (N_NODES / 16) * (DOUT / 16);
    k_gemm_wmma<DIN, DOUT><<<cdiv((long long)waves * 32, TPB), TPB, 0, stream>>>(xin, lin, bufH);
  }
  k_node_scores<DOUT><<<cdiv((long long)N_NODES * 32, TPB), TPB, 0, stream>>>(
      bufH, att_src, att_dst, as_, ad_);
  k_zero<<<cdiv(N_NODES, TPB), TPB, 0, stream>>>((float*)menc, N_NODES);  // 0 == enc(min)
  k_zero<<<cdiv(N_NODES, TPB), TPB, 0, stream>>>(ssum, N_NODES);
  k_zero4<<<cdiv((long long)N_NODES * DOUT / 4, TPB), TPB, 0, stream>>>(
      (float4*)out, (long long)N_NODES * DOUT / 4);
  k_edge_score_max<<<cdiv(ETOT, TPB), TPB, 0, stream>>>(ei, ew, scal, as_, ad_, aex, menc);
  k_edge_exp_sum<<<cdiv(ETOT, TPB), TPB, 0, stream>>>(ei, aex, menc, ssum);
  k_edge_norm<<<cdiv(ETOT, TPB), TPB, 0, stream>>>(ei, aex, ssum);
  k_aggregate<DOUT><<<cdiv((long long)ETOT * (DOUT / 4), TPB), TPB, 0, stream>>>(ei, aex, bufH,
                                                                                 out);
  k_bias_act<DOUT><<<cdiv((long long)N_NODES * (DOUT / 4), TPB), TPB, 0, stream>>>(out, bias,
                                                                                   relu);
}

template <int D>
static void run_bn(float* x, const float* gamma, const float* beta, float* bnsum,
                   hipStream_t stream) {
  k_zero<<<1, 128, 0, stream>>>(bnsum, 128);  // bnsum[64] ++ bnsq[64] contiguous
  k_bn_reduce<D><<<240, 256, 0, stream>>>(x, bnsum, bnsum + 64);
  k_bn_apply<D><<<cdiv((long long)N_NODES * (D / 4), 256), 256, 0, stream>>>(x, bnsum,
                                                                             bnsum + 64, gamma,
                                                                             beta);
}

extern "C" void kernel_launch(void* const* d_in, const int* in_sizes, int n_in, void* d_out,
                              int out_size, void* d_ws, size_t ws_size, hipStream_t stream) {
  (void)in_sizes; (void)n_in; (void)out_size; (void)ws_size;
  // --- inputs (JAX pytree order: sorted dict keys) ---
  const float* x = (const float*)d_in[0];
  const int* ei = (const int*)d_in[1];       // [2, E]
  const float* ew = (const float*)d_in[2];   // [E, 1]
  const int* batch = (const int*)d_in[3];
  const float* bn1_beta = (const float*)d_in[4];
  const float* bn1_gamma = (const float*)d_in[5];
  const float* bn2_beta = (const float*)d_in[6];
  const float* bn2_gamma = (const float*)d_in[7];
  const float* bn3_beta = (const float*)d_in[8];
  const float* bn3_gamma = (const float*)d_in[9];
  // conv blocks: att_dst, att_edge, att_src, bias, lin, lin_edge
  const float* c1_adst = (const float*)d_in[10];
  const float* c1_aedge = (const float*)d_in[11];
  const float* c1_asrc = (const float*)d_in[12];
  const float* c1_bias = (const float*)d_in[13];
  const float* c1_lin = (const float*)d_in[14];
  const float* c1_ledge = (const float*)d_in[15];
  const float* c2_adst = (const float*)d_in[16];
  const float* c2_aedge = (const float*)d_in[17];
  const float* c2_asrc = (const float*)d_in[18];
  const float* c2_bias = (const float*)d_in[19];
  const float* c2_lin = (const float*)d_in[20];
  const float* c2_ledge = (const float*)d_in[21];
  const float* c3_adst = (const float*)d_in[22];
  const float* c3_aedge = (const float*)d_in[23];
  const float* c3_asrc = (const float*)d_in[24];
  const float* c3_bias = (const float*)d_in[25];
  const float* c3_lin = (const float*)d_in[26];
  const float* c3_ledge = (const float*)d_in[27];
  const float* clv_adst = (const float*)d_in[28];  // conv_logvar < conv_mu
  const float* clv_aedge = (const float*)d_in[29];
  const float* clv_asrc = (const float*)d_in[30];
  const float* clv_bias = (const float*)d_in[31];
  const float* clv_lin = (const float*)d_in[32];
  const float* clv_ledge = (const float*)d_in[33];
  const float* cmu_adst = (const float*)d_in[34];
  const float* cmu_aedge = (const float*)d_in[35];
  const float* cmu_asrc = (const float*)d_in[36];
  const float* cmu_bias = (const float*)d_in[37];
  const float* cmu_lin = (const float*)d_in[38];
  const float* cmu_ledge = (const float*)d_in[39];
  const float* gn_bias = (const float*)d_in[40];
  const float* gn_mscale = (const float*)d_in[41];
  const float* gn_weight = (const float*)d_in[42];

  // --- workspace layout (floats) ---
  float* W = (float*)d_ws;
  float* bufH = W;                                   // N*64
  float* bufA = bufH + (size_t)N_NODES * 64;         // N*64
  float* bufB = bufA + (size_t)N_NODES * 64;         // N*64
  float* aex = bufB + (size_t)N_NODES * 64;          // ETOT
  float* as_ = aex + ETOT;                           // N
  float* ad_ = as_ + N_NODES;                        // N
  unsigned* menc = (unsigned*)(ad_ + N_NODES);       // N
  float* ssum = (float*)(menc + N_NODES);            // N
  float* scal = ssum + N_NODES;                      // [0]=mean_ew [1]=c (pad to 16)
  float* bnsum = scal + 16;                          // 64 + 64 (sumsq)
  float* cnt = bnsum + 128;                          // G
  float* meanG = cnt + N_GRAPHS;                     // G*64
  float* varG = meanG + N_GRAPHS * 64;               // G*64

  const int TPB = 256;
  // edge-weight mean (self-loop fill value)
  k_zero<<<1, 32, 0, stream>>>(scal, 2);
  k_mean_reduce<<<256, TPB, 0, stream>>>(ew, scal, N_EDGES);

  // conv1 (1->16) + ReLU + BN1
  run_gat<1, 16>(x, c1_lin, c1_asrc, c1_adst, c1_ledge, c1_aedge, c1_bias, 1, ei, ew, bufA,
                 bufH, aex, as_, ad_, menc, ssum, scal, stream);
  run_bn<16>(bufA, bn1_gamma, bn1_beta, bnsum, stream);
  // conv2 (16->32) + ReLU + BN2
  run_gat<16, 32>(bufA, c2_lin, c2_asrc, c2_adst, c2_ledge, c2_aedge, c2_bias, 1, ei, ew, bufB,
                  bufH, aex, as_, ad_, menc, ssum, scal, stream);
  run_bn<32>(bufB, bn2_gamma, bn2_beta, bnsum, stream);
  // conv3 (32->64) + ReLU + BN3
  run_gat<32, 64>(bufB, c3_lin, c3_asrc, c3_adst, c3_ledge, c3_aedge, c3_bias, 1, ei, ew, bufA,
                  bufH, aex, as_, ad_, menc, ssum, scal, stream);
  run_bn<64>(bufA, bn3_gamma, bn3_beta, bnsum, stream);
  // graph norm: bufA -> (xc in bufB) -> bufA
  k_zero<<<cdiv(N_GRAPHS + 2 * N_GRAPHS * 64, TPB), TPB, 0, stream>>>(
      cnt, N_GRAPHS + 2 * N_GRAPHS * 64);
  k_gn_cnt<<<cdiv(N_NODES, TPB), TPB, 0, stream>>>(batch, cnt);
  k_gn_mean<<<cdiv((long long)N_NODES * 16, TPB), TPB, 0, stream>>>(bufA, batch, meanG);
  k_gn_xcvar<<<cdiv((long long)N_NODES * 16, TPB), TPB, 0, stream>>>(bufA, batch, cnt, meanG,
                                                                     gn_mscale, bufB, varG);
  k_gn_apply<<<cdiv((long long)N_NODES * 16, TPB), TPB, 0, stream>>>(bufB, batch, cnt, varG,
                                                                     gn_weight, gn_bias, bufA);
  // conv_mu / conv_logvar (64->64), no relu, straight into d_out (mu ++ logvar)
  float* outf = (float*)d_out;
  run_gat<64, 64>(bufA, cmu_lin, cmu_asrc, cmu_adst, cmu_ledge, cmu_aedge, cmu_bias, 0, ei, ew,
                  outf, bufH, aex, as_, ad_, menc, ssum, scal, stream);
  run_gat<64, 64>(bufA, clv_lin, clv_asrc, clv_adst, clv_ledge, clv_aedge, clv_bias, 0, ei, ew,
                  outf + (size_t)N_NODES * 64, bufH, aex, as_, ad_, menc, ssum, scal, stream);
}